// YoloLoss_53953379172746
// MI455X (gfx1250) — compile-verified
//
#include <hip/hip_runtime.h>

// YOLO-v1 loss, MI455X (gfx1250, wave32).
// HBM-bound: 193 MB single pass -> b128-staged coalesced streaming into LDS,
// thread-per-cell compute, deterministic two-stage reduction.
// Final cross-block reduction uses V_WMMA_F32_16X16X4_F32 (B = ones) as a
// deterministic 32-lane adder.

typedef __attribute__((ext_vector_type(2))) float v2f;
typedef __attribute__((ext_vector_type(4))) float v4f;
typedef __attribute__((ext_vector_type(8))) float v8f;

#define TILE    256                 // cells per LDS tile
#define NCHAN   30
#define STRIDE  31                  // 30 data + 1 pad; gcd(31,64)=1 -> conflict-free reads
#define CELLS   (4096 * 14 * 14)    // 802816
#define NTILES  (CELLS / TILE)      // 3136 exactly
#define NVEC    (TILE * NCHAN / 4)  // 1920 float4 per tensor per tile
#define NBLOCKS 1024

__launch_bounds__(256)
__global__ void yolo_main(const float* __restrict__ pred,
                          const float* __restrict__ targ,
                          float* __restrict__ partials) {
  __shared__ float sp[TILE * STRIDE];
  __shared__ float st[TILE * STRIDE];
  __shared__ float wsum[8];

  const int tid = threadIdx.x;
  float acc = 0.0f;

  for (int tile = blockIdx.x; tile < NTILES; tile += gridDim.x) {
    const long base = (long)tile * (TILE * NCHAN);
    // Tile base is 30720 B from a 256B-aligned allocation -> 16B-aligned.
    const v4f* __restrict__ gp = reinterpret_cast<const v4f*>(pred + base);
    const v4f* __restrict__ gt = reinterpret_cast<const v4f*>(targ + base);

    // Coalesced b128 global loads; scalar pad-inserting scatter into LDS.
    // LDS addr for flat element e: c*31 + (e - 30c) = e + e/30.
    for (int idx4 = tid; idx4 < NVEC; idx4 += 256) {
      const v4f vp = gp[idx4];
      const v4f vt = gt[idx4];
      const int e0 = idx4 * 4;
      #pragma unroll
      for (int j = 0; j < 4; ++j) {
        const int e = e0 + j;
        const int addr = e + e / NCHAN;
        sp[addr] = vp[j];
        st[addr] = vt[j];
      }
    }
    __syncthreads();

    const float* p = sp + tid * STRIDE;
    const float* t = st + tid * STRIDE;

    const float conf = t[4];
    float cell;
    if (conf > 0.0f) {
      // class loss (channels 10..29)
      float cls = 0.0f;
      #pragma unroll
      for (int c = 10; c < 30; ++c) {
        const float d = p[c] - t[c];
        cls += d * d;
      }
      // target box 0 corners (match reference: areas via corner subtraction;
      // keep IEEE /14.0f so argmax tie behavior matches bit-exactly)
      const float tltx = t[0] / 14.0f - 0.5f * t[2];
      const float tlty = t[1] / 14.0f - 0.5f * t[3];
      const float trbx = t[0] / 14.0f + 0.5f * t[2];
      const float trby = t[1] / 14.0f + 0.5f * t[3];
      const float area_t = (trbx - tltx) * (trby - tlty);

      float iou0 = 0.0f, iou1 = 0.0f;
      #pragma unroll
      for (int b = 0; b < 2; ++b) {
        const float* pb = p + b * 5;
        const float pltx = pb[0] / 14.0f - 0.5f * pb[2];
        const float plty = pb[1] / 14.0f - 0.5f * pb[3];
        const float prbx = pb[0] / 14.0f + 0.5f * pb[2];
        const float prby = pb[1] / 14.0f + 0.5f * pb[3];
        const float ltx = fmaxf(pltx, tltx), lty = fmaxf(plty, tlty);
        const float rbx = fminf(prbx, trbx), rby = fminf(prby, trby);
        const float wx = fmaxf(rbx - ltx, 0.0f);
        const float wy = fmaxf(rby - lty, 0.0f);
        const float inter  = wx * wy;
        const float area_p = (prbx - pltx) * (prby - plty);
        const float iou = inter / (area_p + area_t - inter);
        if (b == 0) iou0 = iou; else iou1 = iou;
      }
      const int   best = (iou1 > iou0) ? 1 : 0;   // jnp.argmax tie -> first
      const float miou = fmaxf(iou0, iou1);
      const float* pbr = p + best * 5;
      const float* tbr = t + best * 5;

      const float dcf = pbr[4] - miou;
      const float dx  = pbr[0] - tbr[0];
      const float dy  = pbr[1] - tbr[1];
      const float dw  = sqrtf(pbr[2]) - sqrtf(tbr[2]);
      const float dh  = sqrtf(pbr[3]) - sqrtf(tbr[3]);
      cell = cls + dcf * dcf + 5.0f * (dx * dx + dy * dy + dw * dw + dh * dh);
    } else {
      const float d4 = p[4] - t[4];
      const float d9 = p[9] - t[9];
      cell = 0.5f * (d4 * d4 + d9 * d9);
    }
    acc += cell;
    __syncthreads();   // protect LDS before next tile overwrites it
  }

  // wave32 shuffle-tree reduction (deterministic order)
  #pragma unroll
  for (int off = 16; off > 0; off >>= 1) acc += __shfl_xor(acc, off, 32);
  if ((tid & 31) == 0) wsum[tid >> 5] = acc;
  __syncthreads();
  if (tid == 0) {
    float b = 0.0f;
    #pragma unroll
    for (int w = 0; w < 8; ++w) b += wsum[w];
    partials[blockIdx.x] = b;
  }
}

// One wave (32 lanes, EXEC all-1s at the WMMA). Each lane strided-sums the
// block partials; the 32 lane accumulators are summed by a single
// v_wmma_f32_16x16x4_f32 against an all-ones B matrix: D[m][n] = rowsum_m,
// and sum_m D[m][0] = sum of every A element = sum of all 32 accumulators.
// C layout: lane 0 holds D[0..7][0] in c[0..7], lane 16 holds D[8..15][0].
__global__ void yolo_finalize(const float* __restrict__ partials,
                              float* __restrict__ out) {
  const int lane = threadIdx.x;
  float acc = 0.0f;
  for (int i = lane; i < NBLOCKS; i += 32) acc += partials[i];  // uniform count

  v2f a;  a[0] = acc;  a[1] = 0.0f;
  v2f bo; bo[0] = 1.0f; bo[1] = 1.0f;
  v8f c = {};
  c = __builtin_amdgcn_wmma_f32_16x16x4_f32(
      /*neg_a=*/false, a, /*neg_b=*/false, bo,
      /*c_mod=*/(short)0, c, /*reuse_a=*/false, /*reuse_b=*/false);

  float s = c[0] + c[1] + c[2] + c[3] + c[4] + c[5] + c[6] + c[7];
  const float other = __shfl(s, 16, 32);     // lane 16's half (M=8..15)
  if (lane == 0) out[0] = (s + other) * (1.0f / 4096.0f);  // / BATCH, exact
}

extern "C" void kernel_launch(void* const* d_in, const int* in_sizes, int n_in,
                              void* d_out, int out_size, void* d_ws, size_t ws_size,
                              hipStream_t stream) {
  const float* pred = (const float*)d_in[0];
  const float* targ = (const float*)d_in[1];
  float* partials   = (float*)d_ws;          // NBLOCKS floats of scratch
  float* out        = (float*)d_out;

  yolo_main<<<NBLOCKS, 256, 0, stream>>>(pred, targ, partials);
  yolo_finalize<<<1, 32, 0, stream>>>(partials, out);
}